// GraphConv_88699664597650
// MI455X (gfx1250) — compile-verified
//
#include <hip/hip_runtime.h>
#include <hip/hip_bf16.h>

typedef float v2f __attribute__((ext_vector_type(2)));
typedef float v8f __attribute__((ext_vector_type(8)));

#define D 64  // D_IN == D_OUT == 64

// ---------------------------------------------------------------------------
// 1) Edge-count degrees: atomicAdd 1.0f per edge endpoint.
// ---------------------------------------------------------------------------
__global__ void degree_kernel(const int* __restrict__ src, const int* __restrict__ dst,
                              float* __restrict__ out_deg, float* __restrict__ in_deg,
                              int nE) {
    int e = blockIdx.x * blockDim.x + threadIdx.x;
    if (e < nE) {
        atomicAdd(&out_deg[src[e]], 1.0f);
        atomicAdd(&in_deg[dst[e]], 1.0f);
    }
}

// ---------------------------------------------------------------------------
// 2) In-place deg -> 1/sqrt(max(deg,1)). Covers out_deg and in_deg (contiguous).
// ---------------------------------------------------------------------------
__global__ void norm_kernel(float* __restrict__ buf, int n) {
    int i = blockIdx.x * blockDim.x + threadIdx.x;
    if (i < n) {
        float d = buf[i];
        d = d < 1.0f ? 1.0f : d;
        buf[i] = 1.0f / sqrtf(d);
    }
}

// ---------------------------------------------------------------------------
// 3) h = (feat @ W^T + b) * out_norm[row], using V_WMMA_F32_16X16X4_F32.
//    Block = 128 threads = 4 wave32s. Each block computes a 16-row stripe;
//    wave w computes the 16-column tile starting at w*16. K=64 -> 16 chained
//    accumulating WMMAs per tile.
//
//    A frag (16x4 f32, 2 VGPRs): lanes 0-15 hold row M=lane, K = k0+{0,1};
//                                lanes 16-31 hold row M=lane-16, K = k0+{2,3}.
//    B frag (4x16 f32, 2 VGPRs): lanes 0-15 hold K=k0+{0,1}, N = lane;
//                                lanes 16-31 hold K=k0+{2,3}, N = lane-16.
//    C/D (16x16 f32, 8 VGPRs):   VGPR r = row (r + 8*(lane>=16)), col = lane&15.
// ---------------------------------------------------------------------------
__global__ void gemm_wmma_kernel(const float* __restrict__ feat,
                                 const float* __restrict__ W,
                                 const float* __restrict__ bias,
                                 const float* __restrict__ out_norm,
                                 float* __restrict__ h) {
    const int lane = threadIdx.x & 31;
    const int wave = threadIdx.x >> 5;          // 0..3 -> column tile
    const int lh   = lane >> 4;                 // lane-half: 0 or 1
    const int ll   = lane & 15;

    const int row_base = blockIdx.x * 16;       // 3125 blocks * 16 = 50000 exactly
    const int col_base = wave * 16;

    const int rowA = row_base + ll;             // feat row this lane supplies
    const int colB = col_base + ll;             // W row (= B column) this lane supplies

    const float* __restrict__ Aptr = feat + (size_t)rowA * D + lh * 2;
    const float* __restrict__ Bptr = W    + (size_t)colB * D + lh * 2;

    v8f c = {};
#pragma unroll
    for (int k0 = 0; k0 < D; k0 += 4) {
        v2f a = *(const v2f*)(Aptr + k0);       // feat[rowA][k0 + lh*2 + {0,1}]
        v2f b = *(const v2f*)(Bptr + k0);       // W[colB][k0 + lh*2 + {0,1}] = B^T
        c = __builtin_amdgcn_wmma_f32_16x16x4_f32(
                /*neg_a=*/false, a, /*neg_b=*/false, b,
                /*c_mod=*/(short)0, c, /*reuse_a=*/false, /*reuse_b=*/false);
    }

    const float bv = bias[colB];                // column of this lane in C/D
#pragma unroll
    for (int r = 0; r < 8; ++r) {
        const int row = row_base + r + lh * 8;
        h[(size_t)row * D + col_base + ll] = (c[r] + bv) * out_norm[row];
    }
}

// ---------------------------------------------------------------------------
// 4) Scatter-add: out[dst] += h[src] * in_norm[dst]. 64 threads per edge
//    (one per feature) -> fully coalesced 256B gathers, f32 atomics at L2.
// ---------------------------------------------------------------------------
__global__ void scatter_kernel(const int* __restrict__ src, const int* __restrict__ dst,
                               const float* __restrict__ h, const float* __restrict__ in_norm,
                               float* __restrict__ out, int nE) {
    long long t = (long long)blockIdx.x * blockDim.x + threadIdx.x;
    int e = (int)(t >> 6);
    int j = (int)(t & 63);
    if (e < nE) {
        int s = src[e];
        int d = dst[e];
        float v = h[(size_t)s * D + j] * in_norm[d];
        atomicAdd(&out[(size_t)d * D + j], v);
    }
}

// ---------------------------------------------------------------------------
// Launch
// ---------------------------------------------------------------------------
extern "C" void kernel_launch(void* const* d_in, const int* in_sizes, int n_in,
                              void* d_out, int out_size, void* d_ws, size_t ws_size,
                              hipStream_t stream) {
    const float* feat = (const float*)d_in[0];   // [N, 64]
    const float* W    = (const float*)d_in[1];   // [64, 64]
    const float* bias = (const float*)d_in[2];   // [64]
    const int*   src  = (const int*)d_in[3];     // [nE]
    const int*   dst  = (const int*)d_in[4];     // [nE]

    const int N  = in_sizes[0] / D;              // 50000
    const int nE = in_sizes[3];                  // 800000

    // Workspace layout: out_norm[N] | in_norm[N] | h[N*D]
    float* out_norm = (float*)d_ws;
    float* in_norm  = out_norm + N;
    float* h        = in_norm + N;
    float* out      = (float*)d_out;

    // Zero degree accumulators and output (graph-capture-safe async memsets).
    hipMemsetAsync(out_norm, 0, (size_t)2 * N * sizeof(float), stream);
    hipMemsetAsync(out, 0, (size_t)out_size * sizeof(float), stream);

    degree_kernel<<<(nE + 255) / 256, 256, 0, stream>>>(src, dst, out_norm, in_norm, nE);
    norm_kernel<<<(2 * N + 255) / 256, 256, 0, stream>>>(out_norm, 2 * N);

    // M = 50000 = 3125 * 16 exactly; 4 waves per block cover the 64 columns.
    gemm_wmma_kernel<<<N / 16, 128, 0, stream>>>(feat, W, bias, out_norm, h);

    long long total = (long long)nE * D;
    scatter_kernel<<<(int)((total + 255) / 256), 256, 0, stream>>>(src, dst, h, in_norm, out, nE);
}